// GenModel_56392920597013
// MI455X (gfx1250) — compile-verified
//
#include <hip/hip_runtime.h>
#include <math.h>

// CDNA5 / gfx1250, wave32. One wave integrates 16 particles with
// V_WMMA_F32_16X16X4_F32; triangular early-exit (particle k needs k steps).
// tanh uses the CDNA5 hardware V_TANH_F32 transcendental (branchless).

typedef __attribute__((ext_vector_type(2))) float v2f;
typedef __attribute__((ext_vector_type(8))) float v8f;

// swap lane halves (lane ^= 16) via LDS lane-permute hardware (no LDS memory)
__device__ __forceinline__ float swap16(float x) {
    int i = __builtin_amdgcn_ds_swizzle(__float_as_int(x), 0x401F); // xor 0x10
    return __int_as_float(i);
}

// Hardware tanh (V_TANH_F32 on gfx1250); branchless fallback via V_EXP_F32.
__device__ __forceinline__ float fast_tanh(float x) {
#if __has_builtin(__builtin_amdgcn_tanhf)
    return __builtin_amdgcn_tanhf(x);
#else
    // tanh(x) = sign(x) * (1 - e) / (1 + e),  e = exp(-2|x|) = exp2(-2|x|*log2 e)
    float ax = __builtin_fabsf(x);
    float e  = __builtin_amdgcn_exp2f(-2.885390081777927f * ax);
    float t  = (1.0f - e) * __builtin_amdgcn_rcpf(1.0f + e);
    return __builtin_copysignf(t, x);
#endif
}

__global__ void __launch_bounds__(32)
sde_scan_wmma(const float* __restrict__ z0, const int* __restrict__ idx,
              const float* __restrict__ W1, const float* __restrict__ b1,
              const float* __restrict__ W2, const float* __restrict__ b2,
              const float* __restrict__ log_noise,
              const float* __restrict__ noise,
              float* __restrict__ out, int N)
{
    const int lane = threadIdx.x;   // 0..31
    const int p    = lane & 15;     // particle sub-index within the 16
    const int g    = lane >> 4;     // lane half
    const bool g0  = (g == 0);
    const int p0   = blockIdx.x * 16;

    const float dt  = 1.0f / (float)(N - 1);
    const float sig = expf(log_noise[0]) * sqrtf(dt);

    // ---- preload weights into WMMA operand layouts (once) ----
    // Layer1 A = W1^T (64x8): tile t (M rows 16t..16t+15), chunk kc (K 4kc..4kc+3)
    // A-layout: lane l holds A[l&15, 2g+b] -> W1[4kc+2g+b][16t + p]
    float a1[4][2][2];
    #pragma unroll
    for (int t = 0; t < 4; ++t)
        #pragma unroll
        for (int kc = 0; kc < 2; ++kc)
            #pragma unroll
            for (int b = 0; b < 2; ++b)
                a1[t][kc][b] = W1[(4*kc + 2*g + b) * 64 + 16*t + p];

    // Layer2 A = W2^T (8x64, padded to 16 rows): 16 K-chunks
    float a2[16][2];
    #pragma unroll
    for (int kc = 0; kc < 16; ++kc)
        #pragma unroll
        for (int b = 0; b < 2; ++b)
            a2[kc][b] = (p < 8) ? W2[(4*kc + 2*g + b) * 8 + p] : 0.0f;

    // bias1 in C-layout of H^T: hb[t][v] = b1[16t + 8g + v]
    float hb[4][8];
    #pragma unroll
    for (int t = 0; t < 4; ++t)
        #pragma unroll
        for (int v = 0; v < 8; ++v)
            hb[t][v] = b1[16*t + 8*g + v];

    // bias2 for the 4 state dims this lane owns: dims 2g,2g+1,2g+4,2g+5
    float bb2[4] = { b2[2*g], b2[2*g + 1], b2[2*g + 4], b2[2*g + 5] };

    // ---- initial state: lane l holds Y[p0+p][2g],[2g+1],[2g+4],[2g+5] ----
    const int zi = idx[p0 + p];
    float y0a = z0[zi*8 + 2*g + 0];
    float y0b = z0[zi*8 + 2*g + 1];
    float y1a = z0[zi*8 + 2*g + 4];
    float y1b = z0[zi*8 + 2*g + 5];

    // out[0] = initial state of particle 0 (wave 0 only)
    if (p0 == 0 && p == 0) {
        out[2*g + 0] = y0a; out[2*g + 1] = y0b;
        out[2*g + 4] = y1a; out[2*g + 5] = y1b;
    }

    const int kmax = p0 + 15;                       // last step this wave needs
    const float* nz = noise + (size_t)(p0 + p) * 8; // lane's noise base (step 1)
    const size_t stepStride = (size_t)N * 8;

    // software-pipelined noise load (step k=1)
    v2f e0 = *(const v2f*)(nz + 2*g);
    v2f e1 = *(const v2f*)(nz + 2*g + 4);

    for (int k = 1; k <= kmax; ++k) {
        v2f ec0 = e0, ec1 = e1;

        // issue next step's loads early; clamp index -> no branch in loop body
        {
            const int kn = (k < kmax) ? k : (kmax - 1);
            const float* nn = nz + (size_t)kn * stepStride;
            e0 = *(const v2f*)(nn + 2*g);
            e1 = *(const v2f*)(nn + 2*g + 4);
            const int kp = (k + 3 < kmax) ? (k + 3) : kn;   // HBM prefetch ahead
            __builtin_prefetch(nz + (size_t)kp * stepStride, 0, 1);
        }

        // ---------- layer 1: H^T = W1^T @ Y^T + b1 (8 WMMAs) ----------
        v2f ya = { y0a, y0b };                       // B chunk0 (K=0..3)
        v2f yb = { y1a, y1b };                       // B chunk1 (K=4..7)
        float h[4][8];
        #pragma unroll
        for (int t = 0; t < 4; ++t) {
            v8f c;
            #pragma unroll
            for (int v = 0; v < 8; ++v) c[v] = hb[t][v];
            v2f A0 = { a1[t][0][0], a1[t][0][1] };
            v2f A1 = { a1[t][1][0], a1[t][1][1] };
            c = __builtin_amdgcn_wmma_f32_16x16x4_f32(false, A0, false, ya,
                                                      (short)0, c, false, false);
            c = __builtin_amdgcn_wmma_f32_16x16x4_f32(false, A1, false, yb,
                                                      (short)0, c, false, false);
            #pragma unroll
            for (int v = 0; v < 8; ++v) h[t][v] = fast_tanh(c[v]);
        }

        // ---------- layer 2: D^T = W2^T @ H^T (16 WMMAs, 2 accumulators) ----------
        // C-layout of H^T -> B-layout needs a lane-half swap for half the values.
        v8f d0 = { 0,0,0,0,0,0,0,0 };
        v8f d1 = { 0,0,0,0,0,0,0,0 };
        #pragma unroll
        for (int t = 0; t < 4; ++t) {
            float s[8];
            #pragma unroll
            for (int v = 0; v < 8; ++v) s[v] = swap16(h[t][v]);
            #pragma unroll
            for (int q = 0; q < 4; ++q) {
                const int kc = 4*t + q;
                float bv0, bv1;
                if (q < 2) {            // local for g=0, remote for g=1
                    bv0 = g0 ? h[t][4*q + 0] : s[4*q + 2];
                    bv1 = g0 ? h[t][4*q + 1] : s[4*q + 3];
                } else {                // local for g=1, remote for g=0
                    const int r = q - 2;
                    bv0 = g0 ? s[4*r + 0] : h[t][4*r + 2];
                    bv1 = g0 ? s[4*r + 1] : h[t][4*r + 3];
                }
                v2f A = { a2[kc][0], a2[kc][1] };
                v2f B = { bv0, bv1 };
                if (kc & 1)
                    d1 = __builtin_amdgcn_wmma_f32_16x16x4_f32(false, A, false, B,
                                                               (short)0, d1, false, false);
                else
                    d0 = __builtin_amdgcn_wmma_f32_16x16x4_f32(false, A, false, B,
                                                               (short)0, d0, false, false);
            }
        }
        v8f d;
        #pragma unroll
        for (int v = 0; v < 8; ++v) d[v] = d0[v] + d1[v];

        // ---- gather this lane's 4 drift dims (g=0 rows hold all 8 dims) ----
        float u0 = g0 ? d[0] : swap16(d[2]);
        float u1 = g0 ? d[1] : swap16(d[3]);
        float u2 = g0 ? d[4] : swap16(d[6]);
        float u3 = g0 ? d[5] : swap16(d[7]);
        u0 += bb2[0]; u1 += bb2[1]; u2 += bb2[2]; u3 += bb2[3];

        // ---- Euler–Maruyama update ----
        y0a += dt * u0 + sig * ec0.x;
        y0b += dt * u1 + sig * ec0.y;
        y1a += dt * u2 + sig * ec1.x;
        y1b += dt * u3 + sig * ec1.y;

        // ---- record particle k at step k ----
        if (k >= p0 && (k - p0) == p) {
            float* o = out + (size_t)k * 8;
            o[2*g + 0] = y0a; o[2*g + 1] = y0b;
            o[2*g + 4] = y1a; o[2*g + 5] = y1b;
        }
    }
}

extern "C" void kernel_launch(void* const* d_in, const int* in_sizes, int n_in,
                              void* d_out, int out_size, void* d_ws, size_t ws_size,
                              hipStream_t stream) {
    const float* z0        = (const float*)d_in[0];
    const int*   idx       = (const int*)  d_in[1];
    const float* W1        = (const float*)d_in[2];
    const float* b1        = (const float*)d_in[3];
    const float* W2        = (const float*)d_in[4];
    const float* b2        = (const float*)d_in[5];
    const float* log_noise = (const float*)d_in[6];
    const float* noise     = (const float*)d_in[7];
    float* out = (float*)d_out;

    const int N = in_sizes[1];          // idx has N entries (4096)
    const int waves = N / 16;           // one wave32 per 16 particles

    hipLaunchKernelGGL(sde_scan_wmma, dim3(waves), dim3(32), 0, stream,
                       z0, idx, W1, b1, W2, b2, log_noise, noise, out, N);
}